// GATHeadLayer_68101001445814
// MI455X (gfx1250) — compile-verified
//
#include <hip/hip_runtime.h>

#define N_NODES 30000
#define N_EDGES 480000
#define DIM     128
#define LSTR    132                       // padded LDS row stride (floats): 132*4 B, 16B-aligned, bank-spread
#define ZBYTES  ((size_t)N_NODES * DIM * sizeof(float))

typedef __attribute__((ext_vector_type(2))) float v2f;
typedef __attribute__((ext_vector_type(8))) float v8f;

// ---------------- order-preserving float <-> uint (for atomic segment-max) ----
__device__ __forceinline__ unsigned ordf(float f) {
  unsigned u = __float_as_uint(f);
  return (u & 0x80000000u) ? ~u : (u | 0x80000000u);
}
__device__ __forceinline__ float unordf(unsigned u) {
  return (u & 0x80000000u) ? __uint_as_float(u & 0x7fffffffu) : __uint_as_float(~u);
}

// ---------------- z = h @ W^T via V_WMMA_F32_16X16X4_F32 ----------------------
// Block: 256 threads = 8 waves. Block computes a 16-row x 128-col tile of z.
// Wave w computes the 16x16 tile at columns [16w, 16w+16).
__global__ __launch_bounds__(256) void gemm_wmma_f32(const float* __restrict__ h,
                                                     const float* __restrict__ W,
                                                     float* __restrict__ z) {
  __shared__ float sW[DIM * LSTR];   // W[o][k] at sW[o*LSTR + k]
  __shared__ float sH[16 * LSTR];    // h-tile[m][k]
  const int t  = threadIdx.x;
  const int m0 = blockIdx.x * 16;

  // cooperative load of W (128x128) as float4
  for (int i = t; i < DIM * DIM / 4; i += 256) {
    const int row = i >> 5;            // 32 float4 per row
    const int col = (i & 31) * 4;
    *(float4*)&sW[row * LSTR + col] = *(const float4*)&W[row * DIM + col];
  }
  // cooperative load of h tile (16x128)
  for (int i = t; i < 16 * DIM / 4; i += 256) {
    const int row = i >> 5;
    const int col = (i & 31) * 4;
    *(float4*)&sH[row * LSTR + col] = *(const float4*)&h[(size_t)(m0 + row) * DIM + col];
  }
  __syncthreads();

  const int wave = t >> 5;
  const int lane = t & 31;
  const int half = lane >> 4;   // 0 or 1
  const int idx  = lane & 15;
  const int n0   = wave * 16;

  // A 16x4 fragment: lane(idx) = row M, VGPR v holds K = 2*half + v
  // B 4x16 fragment: lane(idx) = col N, VGPR v holds K = 2*half + v  (B[k][n] = W[n][k])
  const float* aRow = &sH[idx * LSTR + 2 * half];
  const float* bRow = &sW[(n0 + idx) * LSTR + 2 * half];

  v8f c = {};   // accumulator, 16x16 f32
#pragma unroll
  for (int k = 0; k < DIM; k += 4) {
    v2f a = { aRow[k], aRow[k + 1] };
    v2f b = { bRow[k], bRow[k + 1] };
    c = __builtin_amdgcn_wmma_f32_16x16x4_f32(false, a, false, b, (short)0, c, false, false);
  }

  // C/D layout: VGPR v -> M = v + 8*half, N = idx
#pragma unroll
  for (int v = 0; v < 8; ++v) {
    z[(size_t)(m0 + v + 8 * half) * DIM + (n0 + idx)] = c[v];
  }
}

// ---------------- edge pass 1: per-channel segment max ------------------------
// One wave per edge; each lane owns 4 channels (32 lanes x 4 = 128).
__global__ __launch_bounds__(256) void edge_max_k(const float* __restrict__ z,
                                                  const int* __restrict__ src,
                                                  const int* __restrict__ dst,
                                                  unsigned* __restrict__ m) {
  const unsigned tid = blockIdx.x * 256u + threadIdx.x;
  const int e = tid >> 5;
  const int c = (tid & 31) * 4;
  const int s = src[e], d = dst[e];           // wave-uniform -> scalar loads
  const float4 zs = *(const float4*)(z + (size_t)s * DIM + c);
  const float4 zd = *(const float4*)(z + (size_t)d * DIM + c);
  unsigned* mp = m + (size_t)d * DIM + c;
  atomicMax(mp + 0, ordf(zs.x * zd.x));
  atomicMax(mp + 1, ordf(zs.y * zd.y));
  atomicMax(mp + 2, ordf(zs.z * zd.z));
  atomicMax(mp + 3, ordf(zs.w * zd.w));
}

// ---------------- edge pass 2: denom = segment_sum(exp(e - m)) ----------------
__global__ __launch_bounds__(256) void edge_sum_k(const float* __restrict__ z,
                                                  const int* __restrict__ src,
                                                  const int* __restrict__ dst,
                                                  const unsigned* __restrict__ m,
                                                  float* __restrict__ denom) {
  const unsigned tid = blockIdx.x * 256u + threadIdx.x;
  const int e = tid >> 5;
  const int c = (tid & 31) * 4;
  const int s = src[e], d = dst[e];
  const float4 zs = *(const float4*)(z + (size_t)s * DIM + c);
  const float4 zd = *(const float4*)(z + (size_t)d * DIM + c);
  const unsigned* mp = m + (size_t)d * DIM + c;
  float* dp = denom + (size_t)d * DIM + c;
  atomicAdd(dp + 0, __expf(zs.x * zd.x - unordf(mp[0])));
  atomicAdd(dp + 1, __expf(zs.y * zd.y - unordf(mp[1])));
  atomicAdd(dp + 2, __expf(zs.z * zd.z - unordf(mp[2])));
  atomicAdd(dp + 3, __expf(zs.w * zd.w - unordf(mp[3])));
}

// ---------------- edge pass 3: hout += alpha * z_src --------------------------
__global__ __launch_bounds__(256) void edge_scatter_k(const float* __restrict__ z,
                                                      const int* __restrict__ src,
                                                      const int* __restrict__ dst,
                                                      const unsigned* __restrict__ m,
                                                      const float* __restrict__ denom,
                                                      float* __restrict__ hout) {
  const unsigned tid = blockIdx.x * 256u + threadIdx.x;
  const int e = tid >> 5;
  const int c = (tid & 31) * 4;
  const int s = src[e], d = dst[e];
  const float4 zs = *(const float4*)(z + (size_t)s * DIM + c);
  const float4 zd = *(const float4*)(z + (size_t)d * DIM + c);
  const unsigned* mp = m + (size_t)d * DIM + c;
  const float4 dn = *(const float4*)(denom + (size_t)d * DIM + c);
  float* hp = hout + (size_t)d * DIM + c;
  atomicAdd(hp + 0, (__expf(zs.x * zd.x - unordf(mp[0])) / dn.x) * zs.x);
  atomicAdd(hp + 1, (__expf(zs.y * zd.y - unordf(mp[1])) / dn.y) * zs.y);
  atomicAdd(hp + 2, (__expf(zs.z * zd.z - unordf(mp[2])) / dn.z) * zs.z);
  atomicAdd(hp + 3, (__expf(zs.w * zd.w - unordf(mp[3])) / dn.w) * zs.w);
}

// ---------------- graph-norm scale + per-channel sum / sumsq ------------------
// 240 blocks x 125 rows. Thread tid owns channel (tid & 127); two threads per
// channel cover alternate rows; LDS-combine then 2 atomics per channel.
__global__ __launch_bounds__(256) void stats_k(float* __restrict__ hout,
                                               const float* __restrict__ snorm,
                                               float* __restrict__ stats) {
  const int c   = threadIdx.x & 127;
  const int sub = threadIdx.x >> 7;
  const int r0  = blockIdx.x * 125;
  float s = 0.f, ss = 0.f;
  for (int r = r0 + sub; r < r0 + 125; r += 2) {
    const size_t i = (size_t)r * DIM + c;
    const float v = hout[i] * snorm[r];
    hout[i] = v;
    s += v;
    ss += v * v;
  }
  __shared__ float red[512];
  red[threadIdx.x]       = s;
  red[256 + threadIdx.x] = ss;
  __syncthreads();
  if (sub == 0) {
    atomicAdd(&stats[c],       s  + red[threadIdx.x + 128]);
    atomicAdd(&stats[128 + c], ss + red[256 + threadIdx.x + 128]);
  }
}

// ---------------- batch-norm + ELU -------------------------------------------
__global__ __launch_bounds__(256) void finalize_k(float* __restrict__ out,
                                                  const float* __restrict__ stats,
                                                  const float* __restrict__ gamma,
                                                  const float* __restrict__ beta) {
  const int i = blockIdx.x * 256 + threadIdx.x;   // exactly N*DIM work items
  const int c = i & 127;
  const float mu  = stats[c] * (1.0f / N_NODES);
  const float var = stats[128 + c] * (1.0f / N_NODES) - mu * mu;
  const float x = (out[i] - mu) * rsqrtf(var + 1e-5f) * gamma[c] + beta[c];
  out[i] = x > 0.f ? x : (__expf(x) - 1.f);
}

extern "C" void kernel_launch(void* const* d_in, const int* in_sizes, int n_in,
                              void* d_out, int out_size, void* d_ws, size_t ws_size,
                              hipStream_t stream) {
  const float* h     = (const float*)d_in[0];
  const float* snorm = (const float*)d_in[1];
  const float* W     = (const float*)d_in[2];
  const float* gamma = (const float*)d_in[3];
  const float* beta  = (const float*)d_in[4];
  const int*   src   = (const int*)d_in[5];
  const int*   dst   = (const int*)d_in[6];
  float* out = (float*)d_out;

  char* ws = (char*)d_ws;
  float*    z     = (float*)(ws);                 // [N, 128] fp32
  unsigned* m     = (unsigned*)(ws + ZBYTES);     // [N, 128] ordered-uint max
  float*    denom = (float*)(ws + 2 * ZBYTES);    // [N, 128]
  float*    stats = (float*)(ws + 3 * ZBYTES);    // [256] sum | sumsq

  hipMemsetAsync(m,     0, ZBYTES, stream);       // 0 == ordered(-huge)
  hipMemsetAsync(denom, 0, ZBYTES, stream);
  hipMemsetAsync(stats, 0, 256 * sizeof(float), stream);
  hipMemsetAsync(out,   0, ZBYTES, stream);       // hout accumulates in d_out

  gemm_wmma_f32 <<<N_NODES / 16,      256, 0, stream>>>(h, W, z);
  edge_max_k    <<<N_EDGES * 32 / 256, 256, 0, stream>>>(z, src, dst, m);
  edge_sum_k    <<<N_EDGES * 32 / 256, 256, 0, stream>>>(z, src, dst, m, denom);
  edge_scatter_k<<<N_EDGES * 32 / 256, 256, 0, stream>>>(z, src, dst, m, denom, out);
  stats_k       <<<N_NODES / 125,     256, 0, stream>>>(out, snorm, stats);
  finalize_k    <<<N_NODES * DIM / 256, 256, 0, stream>>>(out, stats, gamma, beta);
}